// BoltzmannMachine_10703058501754
// MI455X (gfx1250) — compile-verified
//
#include <hip/hip_runtime.h>
#include <math.h>

// Boltzmann machine recurrence, MI455X (gfx1250), fp32 WMMA path, v2.
//
// 512 strictly sequential matvec steps; W (16MB) is L2-resident (192MB L2)
// -> per-step latency bound. v2 changes vs v1:
//  * fold constant x-block into bias b = W[512:,0:512]@x  (K: 2048 -> 1536)
//  * 4-way K-split per 16-row group: 96 blocks x 4 waves = 384 waves/step
//    (4x occupancy, per-wave WMMA chain 512 -> 96 chunks)
//  * 96-entry norm partials tree-reduced in LDS (deterministic)
// Core MAC engine unchanged: V_WMMA_F32_16X16X4_F32 with A = broadcast act
// chunk, B = 16 W-rows (coalesced b64/lane), C/D chained accumulation.

#define IN_    512
#define OUT_   512
#define HID_   1024
#define LAYER  2048
#define ACTN   1536           // active slice: rows/cols 512..2047
#define NITER  512
#define TPB    128            // 4 waves per block
#define KW     384            // K-range per wave (4 waves * 384 = 1536)
#define NBLK   96             // 96 blocks * 16 rows = 1536 active rows
#define EPS_   1e-12f

typedef __attribute__((ext_vector_type(2))) float v2f;
typedef __attribute__((ext_vector_type(8))) float v8f;

__global__ void bm_init(float* __restrict__ actA, float* __restrict__ actB,
                        float* __restrict__ PA, float* __restrict__ PB) {
    int i = blockIdx.x * blockDim.x + threadIdx.x;
    if (i < ACTN) { actA[i] = 0.0f; actB[i] = 0.0f; }   // act0 active slice = 0
    if (i < NBLK) { PA[i] = 0.0f; PB[i] = 0.0f; }       // ||h0|| = 0
}

// One-time: bias[j] = sum_{k<512} W[512+j][k] * x[k]
__global__ void bm_bias(const float* __restrict__ W, const float* __restrict__ x,
                        float* __restrict__ bias) {
    __shared__ float xs[IN_];
    int tid = threadIdx.x;
    for (int j = tid; j < IN_; j += blockDim.x) xs[j] = x[j];
    __syncthreads();
    int j = blockIdx.x * blockDim.x + tid;
    if (j < ACTN) {
        const float* wr = W + (size_t)(IN_ + j) * LAYER;
        float s = 0.0f;
        #pragma unroll 8
        for (int k = 0; k < IN_; ++k) s += wr[k] * xs[k];
        bias[j] = s;
    }
}

__global__ __launch_bounds__(TPB)
void bm_iter(const float* __restrict__ W, const float* __restrict__ bias,
             const float* __restrict__ act_in, const float* __restrict__ P_in,
             float* __restrict__ act_out,      float* __restrict__ P_out) {
    __shared__ float act_lds[ACTN];   // active slice: [y(512), h_raw(1024)]
    __shared__ float red[TPB];
    __shared__ float part[TPB];

    const int tid = threadIdx.x;

    // Stage active act slice into LDS; tree-reduce previous-step norm partials.
    red[tid] = (tid < NBLK) ? P_in[tid] : 0.0f;
    for (int j = tid; j < ACTN; j += TPB) act_lds[j] = act_in[j];
    __syncthreads();
    for (int s = TPB / 2; s > 0; s >>= 1) {
        if (tid < s) red[tid] += red[tid + s];
        __syncthreads();
    }
    const float sprev = 1.0f / fmaxf(sqrtf(red[0]), EPS_);
    // Pre-scale hidden slice: act_lds[512..1535] *= 1/max(||h||,eps)
    for (int j = tid; j < HID_; j += TPB) act_lds[OUT_ + j] *= sprev;
    __syncthreads();

    // 4 waves split K=1536; each wave does 96 WMMA chunks for the block's 16 rows.
    const int lane = tid & 31;
    const int wv   = tid >> 5;
    const int n    = lane & 15;        // output-row column of B / D
    const int half = lane >> 4;        // K split: half 0 -> K0,K1 ; half 1 -> K2,K3
    const int grow = IN_ + blockIdx.x * 16 + n;          // 512..2047
    const float* wp = W + (size_t)grow * LAYER + IN_ + wv * KW + half * 2;
    const float* ap = &act_lds[wv * KW + half * 2];

    v8f c = {};
    #pragma unroll 8
    for (int kk = 0; kk < KW; kk += 4) {
        v2f bv = *(const v2f*)(wp + kk);   // B: 16 W-rows, coalesced b64/lane
        v2f av = *(const v2f*)(ap + kk);   // A: broadcast act chunk (LDS b64)
        c = __builtin_amdgcn_wmma_f32_16x16x4_f32(
                /*neg_a=*/false, av, /*neg_b=*/false, bv,
                /*c_mod=*/(short)0, c, /*reuse_a=*/false, /*reuse_b=*/false);
    }
    part[tid] = c[0];                  // broadcast-A => c[0] == wave's partial dot
    __syncthreads();

    // Combine 4 K-partials + bias, relu, store raw; build norm partial.
    if (tid < 32) {
        float dot = part[tid] + part[tid + 32] + part[tid + 64] + part[tid + 96]
                  + bias[blockIdx.x * 16 + n];
        float val = fmaxf(dot, 0.0f);
        if (half == 0) act_out[blockIdx.x * 16 + n] = val;
        red[tid] = (half == 0 && grow >= IN_ + OUT_) ? val * val : 0.0f;
    } else {
        red[tid] = 0.0f;
    }
    __syncthreads();
    for (int s = TPB / 2; s > 0; s >>= 1) {
        if (tid < s) red[tid] += red[tid + s];
        __syncthreads();
    }
    if (tid == 0) P_out[blockIdx.x] = red[0];
}

__global__ void bm_finish(const float* __restrict__ x,
                          const float* __restrict__ act,
                          const float* __restrict__ P,
                          float* __restrict__ out) {
    __shared__ float s_sh;
    if (threadIdx.x == 0) {
        float s = 0.0f;
        for (int i = 0; i < NBLK; ++i) s += P[i];
        s_sh = 1.0f / fmaxf(sqrtf(s), EPS_);
    }
    __syncthreads();
    const float s = s_sh;
    for (int j = threadIdx.x; j < LAYER; j += blockDim.x) {
        float v;
        if (j < IN_)            v = x[j];
        else if (j < IN_ + OUT_) v = act[j - IN_];
        else                     v = act[j - IN_] * s;   // apply final hidden norm
        out[j] = v;
    }
}

extern "C" void kernel_launch(void* const* d_in, const int* in_sizes, int n_in,
                              void* d_out, int out_size, void* d_ws, size_t ws_size,
                              hipStream_t stream) {
    const float* x = (const float*)d_in[0];
    // d_in[1] = y (unused: y-slice of act0 is zeros)
    const float* W = (const float*)d_in[2];
    // d_in[3] = n (= 512, compile-time NITER)

    float* ws = (float*)d_ws;
    float* bufs[2] = { ws, ws + ACTN };                        // ping-pong act
    float* Ps[2]   = { ws + 2 * ACTN, ws + 2 * ACTN + NBLK };  // norm partials
    float* bias    = ws + 2 * ACTN + 2 * NBLK;                 // 1536 floats

    bm_init<<<(ACTN + 255) / 256, 256, 0, stream>>>(bufs[0], bufs[1], Ps[0], Ps[1]);
    bm_bias<<<(ACTN + 255) / 256, 256, 0, stream>>>(W, x, bias);

    for (int t = 0; t < NITER; ++t) {
        bm_iter<<<NBLK, TPB, 0, stream>>>(W, bias,
                                          bufs[t & 1],       Ps[t & 1],
                                          bufs[(t + 1) & 1], Ps[(t + 1) & 1]);
    }
    // After 512 steps the final act/partials land back in buffer 0.
    bm_finish<<<1, 256, 0, stream>>>(x, bufs[0], Ps[0], (float*)d_out);
}